// deepcustNet_76390288327746
// MI455X (gfx1250) — compile-verified
//
#include <hip/hip_runtime.h>

// CDNA5 / gfx1250, wave32.
// Strategy: build CSR (indexed by dst) once per launch, then each GCN layer is
//   gather (pure coalesced reads, no atomics)  ->  fp32 WMMA GEMM + bias + lrelu.
// fp32 WMMA (V_WMMA_F32_16X16X4_F32) is mandatory for precision: no
// normalization in this net => activations grow ~5x/layer to ~1e6 (f16 overflows).
// Compute is negligible (6.6 GFLOP); edge gather traffic (~0.8GB/layer) is the
// roofline term and runs from the 192MB L2 at full bandwidth with zero atomics.

typedef __attribute__((ext_vector_type(2))) float v2f;
typedef __attribute__((ext_vector_type(8))) float v8f;

#define LEAKY_SLOPE 0.01f

__device__ __forceinline__ float lrelu(float v) {
    return v > 0.0f ? v : v * LEAKY_SLOPE;
}

// ========================== CSR construction ================================

__global__ void count_dst(const int* __restrict__ dst, int* __restrict__ counts,
                          int E) {
    int e = blockIdx.x * blockDim.x + threadIdx.x;
    if (e >= E) return;
    atomicAdd(&counts[dst[e]], 1);
}

// Single-block exclusive scan (n ~ 100k): chunked Hillis-Steele with carry.
__global__ void exscan_1block(const int* __restrict__ counts,
                              int* __restrict__ rowptr, int n) {
    __shared__ int sdata[1024];
    __shared__ int carry;
    int tid = threadIdx.x;
    if (tid == 0) carry = 0;
    __syncthreads();
    for (int base = 0; base < n; base += 1024) {
        int i = base + tid;
        int v = (i < n) ? counts[i] : 0;
        sdata[tid] = v;
        __syncthreads();
        for (int off = 1; off < 1024; off <<= 1) {
            int t = (tid >= off) ? sdata[tid - off] : 0;
            __syncthreads();
            sdata[tid] += t;
            __syncthreads();
        }
        int incl = sdata[tid];
        int c = carry;
        if (i < n) rowptr[i] = c + (incl - v);
        __syncthreads();
        if (tid == 1023) carry = c + incl;   // chunk total
        __syncthreads();
    }
    if (tid == 0) rowptr[n] = carry;
}

// Scatter edges into CSR slots: csrsrc[slot] = src[e], bucketed by dst[e].
__global__ void csr_fill(const int* __restrict__ src, const int* __restrict__ dst,
                         int* __restrict__ cursor, int* __restrict__ csrsrc,
                         int E) {
    int e = blockIdx.x * blockDim.x + threadIdx.x;
    if (e >= E) return;
    int slot = atomicAdd(&cursor[dst[e]], 1);
    csrsrc[slot] = src[e];
}

// ========================== CSR gather (pull) ===============================

// 4-dim: one thread per node, float4 accumulator.
__global__ void gather_d4(const float* __restrict__ x,
                          const int* __restrict__ rowptr,
                          const int* __restrict__ csrsrc,
                          float* __restrict__ agg, int N) {
    int n = blockIdx.x * blockDim.x + threadIdx.x;
    if (n >= N) return;
    int beg = rowptr[n], end = rowptr[n + 1];
    float4 acc = {0.f, 0.f, 0.f, 0.f};
    const float4* x4 = reinterpret_cast<const float4*>(x);
    for (int j = beg; j < end; ++j) {
        float4 v = x4[csrsrc[j]];
        acc.x += v.x; acc.y += v.y; acc.z += v.z; acc.w += v.w;
    }
    reinterpret_cast<float4*>(agg)[n] = acc;
}

// 64-dim: one wave per node; lane owns cols [2*lane, 2*lane+1]. Each neighbor
// row is one fully coalesced 256B read (32 lanes x float2). No atomics.
__global__ void gather_d64(const float* __restrict__ x,
                           const int* __restrict__ rowptr,
                           const int* __restrict__ csrsrc,
                           float* __restrict__ agg, int N) {
    int node = blockIdx.x * (blockDim.x >> 5) + (threadIdx.x >> 5);
    if (node >= N) return;
    int lane = threadIdx.x & 31;
    int beg = rowptr[node], end = rowptr[node + 1];
    float2 acc = {0.f, 0.f};
    const float2* x2 = reinterpret_cast<const float2*>(x);
    for (int j = beg; j < end; ++j) {
        int s = csrsrc[j];                       // wave-uniform -> broadcast
        float2 v = x2[(size_t)s * 32 + lane];
        acc.x += v.x; acc.y += v.y;
    }
    reinterpret_cast<float2*>(agg)[(size_t)node * 32 + lane] = acc;
}

// ================= Fallback: atomic scatter (if ws too small) ===============

__global__ void scatter_add_d4(const float* __restrict__ x,
                               const int* __restrict__ src,
                               const int* __restrict__ dst,
                               float* __restrict__ agg, int E) {
    int e = blockIdx.x * blockDim.x + threadIdx.x;
    if (e >= E) return;
    int s = src[e], d = dst[e];
    float4 v = reinterpret_cast<const float4*>(x)[s];
    float* p = agg + (size_t)d * 4;
    unsafeAtomicAdd(p + 0, v.x);
    unsafeAtomicAdd(p + 1, v.y);
    unsafeAtomicAdd(p + 2, v.z);
    unsafeAtomicAdd(p + 3, v.w);
}

__global__ void scatter_add_d64(const float* __restrict__ x,
                                const int* __restrict__ src,
                                const int* __restrict__ dst,
                                float* __restrict__ agg, int E) {
    int t = blockIdx.x * blockDim.x + threadIdx.x;
    int e = t >> 4, q = t & 15;
    if (e >= E) return;
    int s = src[e], d = dst[e];
    float4 v = reinterpret_cast<const float4*>(x + (size_t)s * 64)[q];
    float* p = agg + (size_t)d * 64 + q * 4;
    unsafeAtomicAdd(p + 0, v.x);
    unsafeAtomicAdd(p + 1, v.y);
    unsafeAtomicAdd(p + 2, v.z);
    unsafeAtomicAdd(p + 3, v.w);
}

// ========================== WMMA GEMM layers ================================
// V_WMMA_F32_16X16X4_F32 layouts (wave32):
//   A 16x4 : lane l -> row l&15, VGPR0/1 = K = 2*(l>>4) + {0,1}
//   B 4x16 : lane l -> col l&15, VGPR0: K = 2*(l>>4); VGPR1: K = 2*(l>>4)+1
//   C 16x16: VGPR r -> row r + 8*(l>>4), col l&15

// Layer 1: out[N,64] = lrelu(agg4[N,4] @ W1[4,64] + b1). K=4 => 1 WMMA/tile.
__global__ void gemm_k4_wmma(const float* __restrict__ A,
                             const float* __restrict__ W,
                             const float* __restrict__ bias,
                             float* __restrict__ out, int N) {
    __shared__ float sW[4 * 64];
    __shared__ float sB[64];
    int tid = threadIdx.x;
    sW[tid] = W[tid];                 // 256 threads == 256 weights
    if (tid < 64) sB[tid] = bias[tid];
    __syncthreads();

    int wave = blockIdx.x * (blockDim.x >> 5) + (tid >> 5);
    int row0 = wave * 16;
    if (row0 + 16 > N) return;        // N = 100000 is a multiple of 16

    int lane = tid & 31;
    int half = lane >> 4;
    int ln   = lane & 15;

    const float* arow = A + (size_t)(row0 + ln) * 4 + 2 * half;
    v2f a;
    a.x = arow[0];
    a.y = arow[1];

#pragma unroll
    for (int j = 0; j < 4; ++j) {
        const float* w0 = sW + (2 * half) * 64 + 16 * j + ln;
        v2f b;
        b.x = w0[0];
        b.y = w0[64];
        v8f c = {};
        c = __builtin_amdgcn_wmma_f32_16x16x4_f32(
                false, a, false, b, (short)0, c, false, false);
        float bj = sB[16 * j + ln];
#pragma unroll
        for (int r = 0; r < 8; ++r) {
            float v = c[r] + bj;
            out[(size_t)(row0 + r + 8 * half) * 64 + 16 * j + ln] = lrelu(v);
        }
    }
}

// Layers 2-8: out[N,64] = lrelu(agg[N,64] @ W2[64,64] + b2). W2 in LDS.
__global__ void gemm_k64_wmma(const float* __restrict__ A,
                              const float* __restrict__ W,
                              const float* __restrict__ bias,
                              float* __restrict__ out, int N) {
    __shared__ float sW[64 * 64];
    __shared__ float sB[64];
    int tid = threadIdx.x;
    {
        const float4* W4 = reinterpret_cast<const float4*>(W);
        float4* sW4 = reinterpret_cast<float4*>(sW);
#pragma unroll
        for (int i = 0; i < 4; ++i)
            sW4[tid + 256 * i] = W4[tid + 256 * i];
        if (tid < 64) sB[tid] = bias[tid];
    }
    __syncthreads();

    int wave = blockIdx.x * (blockDim.x >> 5) + (tid >> 5);
    int row0 = wave * 16;
    if (row0 + 16 > N) return;

    int lane = tid & 31;
    int half = lane >> 4;
    int ln   = lane & 15;

    const float* arow = A + (size_t)(row0 + ln) * 64 + 2 * half;

    v8f c0 = {}, c1 = {}, c2 = {}, c3 = {};
#pragma unroll
    for (int k = 0; k < 64; k += 4) {
        v2f a;
        a.x = arow[k];
        a.y = arow[k + 1];
        const float* w0 = sW + (k + 2 * half) * 64 + ln;
        v2f b0, b1, b2, b3;
        b0.x = w0[0];  b0.y = w0[64];
        b1.x = w0[16]; b1.y = w0[80];
        b2.x = w0[32]; b2.y = w0[96];
        b3.x = w0[48]; b3.y = w0[112];
        c0 = __builtin_amdgcn_wmma_f32_16x16x4_f32(false, a, false, b0, (short)0, c0, false, false);
        c1 = __builtin_amdgcn_wmma_f32_16x16x4_f32(false, a, false, b1, (short)0, c1, false, false);
        c2 = __builtin_amdgcn_wmma_f32_16x16x4_f32(false, a, false, b2, (short)0, c2, false, false);
        c3 = __builtin_amdgcn_wmma_f32_16x16x4_f32(false, a, false, b3, (short)0, c3, false, false);
    }

#define GCN_EPILOGUE(cacc, j)                                                  \
    {                                                                          \
        float bj = sB[16 * (j) + ln];                                          \
        _Pragma("unroll")                                                      \
        for (int r = 0; r < 8; ++r) {                                          \
            float v = cacc[r] + bj;                                            \
            out[(size_t)(row0 + r + 8 * half) * 64 + 16 * (j) + ln] = lrelu(v);\
        }                                                                      \
    }
    GCN_EPILOGUE(c0, 0)
    GCN_EPILOGUE(c1, 1)
    GCN_EPILOGUE(c2, 2)
    GCN_EPILOGUE(c3, 3)
#undef GCN_EPILOGUE
}

// Head: out[N,3] = agg[N,64] @ W5[64,3] + b5 (no activation).
__global__ void head_k64_d3(const float* __restrict__ A,
                            const float* __restrict__ W5,
                            const float* __restrict__ b5,
                            float* __restrict__ out, int N) {
    __shared__ float sW[64 * 3];
    int tid = threadIdx.x;
    if (tid < 192) sW[tid] = W5[tid];
    __syncthreads();
    int n = blockIdx.x * blockDim.x + tid;
    if (n >= N) return;
    const float4* arow = reinterpret_cast<const float4*>(A + (size_t)n * 64);
    float a0 = b5[0], a1 = b5[1], a2 = b5[2];
#pragma unroll
    for (int k4 = 0; k4 < 16; ++k4) {
        float4 v = arow[k4];
        const float* w = sW + k4 * 12;
        a0 += v.x * w[0] + v.y * w[3] + v.z * w[6] + v.w * w[9];
        a1 += v.x * w[1] + v.y * w[4] + v.z * w[7] + v.w * w[10];
        a2 += v.x * w[2] + v.y * w[5] + v.z * w[8] + v.w * w[11];
    }
    float* o = out + (size_t)n * 3;
    o[0] = a0; o[1] = a1; o[2] = a2;
}

// ============================================================================
extern "C" void kernel_launch(void* const* d_in, const int* in_sizes, int n_in,
                              void* d_out, int out_size, void* d_ws, size_t ws_size,
                              hipStream_t stream) {
    const float* features = (const float*)d_in[0];   // [N,4]
    const int*   src      = (const int*)d_in[1];     // [E]
    const int*   dst      = (const int*)d_in[2];     // [E]
    const float* W1       = (const float*)d_in[3];   // [4,64]
    const float* b1       = (const float*)d_in[4];   // [64]
    const float* W2       = (const float*)d_in[5];   // [64,64]
    const float* b2       = (const float*)d_in[6];   // [64]
    const float* W5       = (const float*)d_in[7];   // [64,3]
    const float* b5       = (const float*)d_in[8];   // [3]

    const int N = in_sizes[0] / 4;
    const int E = in_sizes[1];
    float* out = (float*)d_out;

    // Workspace layout.
    char* ws = (char*)d_ws;
    float* bufX = (float*)ws;                      ws += (size_t)N * 64 * 4;
    float* bufA = (float*)ws;                      ws += (size_t)N * 64 * 4;
    int* rowptr = (int*)ws;                        ws += (size_t)(N + 1) * 4;
    int* cursor = (int*)ws;                        ws += (size_t)N * 4;
    int* csrsrc = (int*)ws;                        ws += (size_t)E * 4;
    const size_t need = (size_t)(ws - (char*)d_ws);

    const dim3 blk(256);
    const int tiles      = N / 16;
    const int gemmBlocks = (tiles + 7) / 8;          // 8 waves / block
    const int edgeBlocks = (E + 255) / 256;
    const int nodeBlocks = (N + 255) / 256;
    const int waveBlocks = (N + 7) / 8;              // 1 wave / node

    if (ws_size >= need) {
        // -------- CSR build (once per launch; integer atomics only) --------
        hipMemsetAsync(cursor, 0, (size_t)N * 4, stream);
        count_dst<<<edgeBlocks, blk, 0, stream>>>(dst, cursor, E);
        exscan_1block<<<1, 1024, 0, stream>>>(cursor, rowptr, N);
        hipMemcpyAsync(cursor, rowptr, (size_t)N * 4,
                       hipMemcpyDeviceToDevice, stream);
        csr_fill<<<edgeBlocks, blk, 0, stream>>>(src, dst, cursor, csrsrc, E);

        // -------- Layer 1: gather(4) -> WMMA 4->64 + lrelu --------
        gather_d4<<<nodeBlocks, blk, 0, stream>>>(features, rowptr, csrsrc, bufA, N);
        gemm_k4_wmma<<<gemmBlocks, blk, 0, stream>>>(bufA, W1, b1, bufX, N);

        // -------- Layers 2-8: gather(64) -> WMMA 64->64 + lrelu --------
        for (int l = 0; l < 7; ++l) {
            gather_d64<<<waveBlocks, blk, 0, stream>>>(bufX, rowptr, csrsrc, bufA, N);
            gemm_k64_wmma<<<gemmBlocks, blk, 0, stream>>>(bufA, W2, b2, bufX, N);
        }

        // -------- Layer 9: gather(64) -> 64->3 head --------
        gather_d64<<<waveBlocks, blk, 0, stream>>>(bufX, rowptr, csrsrc, bufA, N);
        head_k64_d3<<<nodeBlocks, blk, 0, stream>>>(bufA, W5, b5, out, N);
    } else {
        // -------- Fallback: atomic scatter pipeline --------
        const int scat64Blk = (E * 16 + 255) / 256;
        hipMemsetAsync(bufA, 0, (size_t)N * 4 * sizeof(float), stream);
        scatter_add_d4<<<edgeBlocks, blk, 0, stream>>>(features, src, dst, bufA, E);
        gemm_k4_wmma<<<gemmBlocks, blk, 0, stream>>>(bufA, W1, b1, bufX, N);
        for (int l = 0; l < 7; ++l) {
            hipMemsetAsync(bufA, 0, (size_t)N * 64 * sizeof(float), stream);
            scatter_add_d64<<<scat64Blk, blk, 0, stream>>>(bufX, src, dst, bufA, E);
            gemm_k64_wmma<<<gemmBlocks, blk, 0, stream>>>(bufA, W2, b2, bufX, N);
        }
        hipMemsetAsync(bufA, 0, (size_t)N * 64 * sizeof(float), stream);
        scatter_add_d64<<<scat64Blk, blk, 0, stream>>>(bufX, src, dst, bufA, E);
        head_k64_d3<<<nodeBlocks, blk, 0, stream>>>(bufA, W5, b5, out, N);
    }
}